// RNNScratch_19851338842447
// MI455X (gfx1250) — compile-verified
//
#include <hip/hip_runtime.h>
#include <cstddef>
#include <cstdint>

// ---------------------------------------------------------------------------
// RNN scratch: h_t = tanh(Wxh[tok] + h_{t-1} @ Whh + bh);  out = Hs @ Whq + bq
// B=64, S=512, H=V=1024, f32 throughout (matches reference precision).
// Matrix core path: V_WMMA_F32_16X16X4_F32 (CDNA5 f32 WMMA, wave32).
// Projection staging uses GLOBAL_LOAD_ASYNC_TO_LDS (ASYNCcnt) when available.
// ---------------------------------------------------------------------------

typedef __attribute__((ext_vector_type(2))) float v2f;
typedef __attribute__((ext_vector_type(8))) float v8f;

#define RNN_B 64
#define RNN_S 512
#define RNN_H 1024
#define RNN_V 1024

#if defined(__has_builtin)
#if __has_builtin(__builtin_amdgcn_global_load_async_to_lds_b32) && \
    __has_builtin(__builtin_amdgcn_s_wait_asynccnt)
#define USE_ASYNC_LDS 1
#endif
#endif
#ifndef USE_ASYNC_LDS
#define USE_ASYNC_LDS 0
#endif

#if USE_ASYNC_LDS
typedef __attribute__((address_space(1))) int g_i32;   // global addrspace
typedef __attribute__((address_space(3))) int l_i32;   // LDS addrspace
#endif

// ---------------------------------------------------------------------------
// Step kernel: one time step of the recurrence.
//   grid = 64 blocks (N-tiles over H), block = 128 threads = 4 waves.
//   wave w owns the 16x16 tile (M = 16*w batch rows, N = 16*blockIdx.x hid cols)
//   C += A(16x4) x B(4x16) over K = H in chunks of 4.
// WMMA f32 16x16x4 operand layout (wave32):
//   A: lane half = lane>>4 selects K pair {2*half, 2*half+1}; m = lane&15 row.
//   B: same K split per lane half; n = lane&15 column.
//   C/D: VGPR r -> row M = r + 8*half; col N = lane&15.
// ---------------------------------------------------------------------------
__global__ __launch_bounds__(128) void rnn_step_kernel(
    const int* __restrict__ tok,      // (B, S)
    const float* __restrict__ Wxh,    // (VOCAB=V, H)
    const float* __restrict__ Whh,    // (H, H)
    const float* __restrict__ bh,     // (H)
    float* __restrict__ Hs,           // (S, B, H) workspace
    int s)
{
    const int lane = threadIdx.x & 31;
    const int wave = threadIdx.x >> 5;       // 0..3 -> M tile
    const int m0   = wave * 16;
    const int n0   = blockIdx.x * 16;
    const int half = lane >> 4;              // 0 or 1
    const int mn   = lane & 15;
    const int koff = half * 2;

    v8f acc = {};

    if (s > 0) {
        const float* hprev = Hs + (size_t)(s - 1) * RNN_B * RNN_H;
        const float* aptr  = hprev + (size_t)(m0 + mn) * RNN_H + koff;      // A[m][k..]
        const float* bptr  = Whh + (size_t)koff * RNN_H + (n0 + mn);        // B[k..][n]
        #pragma unroll 4
        for (int k = 0; k < RNN_H; k += 4) {
            v2f a, b;
            a.x = aptr[0];
            a.y = aptr[1];
            b.x = bptr[0];
            b.y = bptr[RNN_H];
            acc = __builtin_amdgcn_wmma_f32_16x16x4_f32(
                /*neg_a=*/false, a, /*neg_b=*/false, b,
                /*c_mod=*/(short)0, acc, /*reuse_a=*/false, /*reuse_b=*/false);
            aptr += 4;
            bptr += 4 * RNN_H;
        }
    }

    // Epilogue: + Wxh[token] row gather + bias, tanh, store Hs[s].
    const float bias = bh[n0 + mn];
    float* hout = Hs + (size_t)s * RNN_B * RNN_H;
    #pragma unroll
    for (int r = 0; r < 8; ++r) {
        const int m = m0 + r + 8 * half;                 // batch row
        const int t = tok[(size_t)m * RNN_S + s];        // token id
        const float xh = Wxh[(size_t)t * RNN_H + (n0 + mn)];
        hout[(size_t)m * RNN_H + (n0 + mn)] = tanhf(acc[r] + xh + bias);
    }
}

// ---------------------------------------------------------------------------
// Projection kernel: out[b][v][s] = sum_h Hs[s][b][h] * Whq[h][v] + bq[v]
// Per block: fixed (b, 16-wide s-tile). Stage Hs[s0:s0+16, b, :] into LDS
// transposed as lds[h][s_local] (64 KB) so WMMA B-reads are conflict-free and
// lane-coalesced. A = Whq^T tiles read directly (L2-resident, lane-coalesced).
// Staging path: per-lane-addressed async B32 memory->LDS copies (transpose
// happens via per-lane LDS destination addresses), tracked with ASYNCcnt.
// C-tile: M = v (16), N = s (16)  =>  D stores are contiguous along s.
//   grid = (S/16, B) = (32, 64), block = 256 threads = 8 waves, 8 v-tiles/wave.
// ---------------------------------------------------------------------------
__global__ __launch_bounds__(256) void rnn_proj_kernel(
    const float* __restrict__ Hs,     // (S, B, H)
    const float* __restrict__ Whq,    // (H, V)
    const float* __restrict__ bq,     // (V)
    float* __restrict__ out)          // (B, V, S)
{
    __shared__ float lds[16 * RNN_H]; // lds[h*16 + s_local], 64 KB

    const int s0 = blockIdx.x * 16;
    const int b  = blockIdx.y;

    const int lane = threadIdx.x & 31;
    const int wave = threadIdx.x >> 5;   // 0..7

#if USE_ASYNC_LDS
    // Async staging with transpose: each wave owns 2 s_local rows; per row,
    // lanes walk contiguous h (128B per wave-instruction on the global side),
    // each lane scatters its element to lds[h][s_local] directly (no VGPR
    // round-trip, tracked by ASYNCcnt).
    #pragma unroll
    for (int r = 0; r < 2; ++r) {
        const int sl = wave * 2 + r;
        const float* grow = Hs + (size_t)(s0 + sl) * RNN_B * RNN_H
                               + (size_t)b * RNN_H;
        for (int hb = 0; hb < RNN_H; hb += 32) {
            const int h = hb + lane;
            __builtin_amdgcn_global_load_async_to_lds_b32(
                (g_i32*)(grow + h),
                (l_i32*)&lds[h * 16 + sl],
                /*offset=*/0, /*cpol=*/0);
        }
    }
    __builtin_amdgcn_s_wait_asynccnt(0);
#else
    // Sync staging with transpose: 16 threads per s_local row, 64 h each.
    {
        const int sl = threadIdx.x >> 4;          // 0..15
        const int hb = (threadIdx.x & 15) * 64;   // h base
        const float* g = Hs + (size_t)(s0 + sl) * RNN_B * RNN_H
                            + (size_t)b * RNN_H + hb;
        #pragma unroll
        for (int j = 0; j < 64; j += 4) {
            const float4 v4 = *(const float4*)(g + j);
            lds[(hb + j + 0) * 16 + sl] = v4.x;
            lds[(hb + j + 1) * 16 + sl] = v4.y;
            lds[(hb + j + 2) * 16 + sl] = v4.z;
            lds[(hb + j + 3) * 16 + sl] = v4.w;
        }
    }
#endif
    __syncthreads();

    const int half = lane >> 4;
    const int mn   = lane & 15;
    const int koff = half * 2;

    for (int t = 0; t < 8; ++t) {
        const int v0 = (wave * 8 + t) * 16;      // v tile base
        v8f acc = {};
        const float* aptr = Whq + (size_t)koff * RNN_V + (v0 + mn); // A[v][k] = Whq[k][v]
        const float* lp   = &lds[koff * 16 + mn];                   // B[k][s] from LDS
        #pragma unroll 4
        for (int k = 0; k < RNN_H; k += 4) {
            v2f a, bb;
            a.x  = aptr[0];
            a.y  = aptr[RNN_V];
            bb.x = lp[0];
            bb.y = lp[16];
            acc = __builtin_amdgcn_wmma_f32_16x16x4_f32(
                false, a, false, bb, (short)0, acc, false, false);
            aptr += 4 * (size_t)RNN_V;
            lp   += 4 * 16;
        }
        #pragma unroll
        for (int r = 0; r < 8; ++r) {
            const int v = v0 + r + 8 * half;
            out[(size_t)b * RNN_V * RNN_S + (size_t)v * RNN_S + (s0 + mn)] =
                acc[r] + bq[v];
        }
    }
}

// ---------------------------------------------------------------------------
extern "C" void kernel_launch(void* const* d_in, const int* in_sizes, int n_in,
                              void* d_out, int out_size, void* d_ws, size_t ws_size,
                              hipStream_t stream) {
    const int*   tok = (const int*)d_in[0];    // (B, S)
    const float* Wxh = (const float*)d_in[1];  // (V, H)
    const float* Whh = (const float*)d_in[2];  // (H, H)
    const float* Whq = (const float*)d_in[3];  // (H, V)
    const float* bh  = (const float*)d_in[4];  // (H)
    const float* bq  = (const float*)d_in[5];  // (V)
    float* out = (float*)d_out;                // out (B,V,S) then H_final (B,H)
    float* Hs  = (float*)d_ws;                 // (S, B, H) f32 = 128 MiB scratch

    // Sequential recurrence: 512 dependent small GEMM+tanh steps.
    for (int s = 0; s < RNN_S; ++s) {
        rnn_step_kernel<<<dim3(RNN_H / 16), 128, 0, stream>>>(
            tok, Wxh, Whh, bh, Hs, s);
    }

    // Output projection (bulk of FLOPs + all the HBM traffic).
    rnn_proj_kernel<<<dim3(RNN_S / 16, RNN_B), 256, 0, stream>>>(
        Hs, Whq, bq, out);

    // H_final = Hs[S-1], appended after out.
    (void)hipMemcpyAsync(out + (size_t)RNN_B * RNN_V * RNN_S,
                         Hs + (size_t)(RNN_S - 1) * RNN_B * RNN_H,
                         (size_t)RNN_B * RNN_H * sizeof(float),
                         hipMemcpyDeviceToDevice, stream);
}